// YOLOLoss_86509231276454
// MI455X (gfx1250) — compile-verified
//
#include <hip/hip_runtime.h>
#include <hip/hip_bf16.h>

typedef __attribute__((ext_vector_type(2))) float v2f;
typedef __attribute__((ext_vector_type(8))) float v8f;

#define BSZ   16
#define NA    3
#define NC    80
#define HH    76
#define WW    76
#define HW    (HH * WW)
#define NCELL (BSZ * NA * HH * WW)
#define NBLK  256
#define NTHR  256
#define NACC  9          // Sx Sy Sw Sh Sconf_m Sconf_nm Scls cnt ncnt
#define EPSF  1e-7f

__device__ __forceinline__ float sigmoidf(float z) {
    return 1.0f / (1.0f + __expf(-z));
}
__device__ __forceinline__ float bcef(float p, float t) {
    p = fminf(fmaxf(p, EPSF), 1.0f - EPSF);
    return -(t * __logf(p) + (1.0f - t) * __logf(1.0f - p));
}

// Reduce 8 waves x 9 accumulators (LDS tile part[8][16]) with two chained
// f32 WMMAs: A(16x4) = wave-partials (row m = accumulator, k = wave),
// B(4x16) = ones  =>  D[m][n] = sum_k A[m][k].
// D layout (16x16 f32): lane n<16 holds rows 0..7 in c[0..7]; lane n+16 rows 8..15.
__device__ __forceinline__ v8f wmma_reduce8(const float part[8][16], int lane) {
    const int m16 = lane & 15;
    const int kh  = lane >> 4;           // 0: K=0,1   1: K=2,3
    v2f b;  b.x = 1.0f; b.y = 1.0f;
    v2f a0; a0.x = part[2 * kh + 0][m16]; a0.y = part[2 * kh + 1][m16];
    v2f a1; a1.x = part[2 * kh + 4][m16]; a1.y = part[2 * kh + 5][m16];
    v8f c = {};
    c = __builtin_amdgcn_wmma_f32_16x16x4_f32(false, a0, false, b, (short)0, c, false, false);
    c = __builtin_amdgcn_wmma_f32_16x16x4_f32(false, a1, false, b, (short)0, c, false, false);
    return c;
}

__global__ __launch_bounds__(NTHR) void yolo_partial_kernel(
    const float* __restrict__ in,
    const unsigned char* __restrict__ mask,
    const unsigned char* __restrict__ nmask,
    const float* __restrict__ tx, const float* __restrict__ ty,
    const float* __restrict__ tw, const float* __restrict__ th,
    const float* __restrict__ tconf,
    const float* __restrict__ tcls,
    const float* __restrict__ bls,
    float* __restrict__ ws)
{
    float acc[NACC];
#pragma unroll
    for (int j = 0; j < NACC; ++j) acc[j] = 0.0f;

    for (int idx = blockIdx.x * NTHR + threadIdx.x; idx < NCELL; idx += NBLK * NTHR) {
        const int i    = idx % WW;
        const int jrow = (idx / WW) % HH;
        const int ba   = idx / HW;                       // b*NA + a
        const float* base = in + (size_t)ba * (5 + NC) * HW + jrow * WW + i;

        const float m  = mask[idx]  ? 1.0f : 0.0f;
        const float nm = nmask[idx] ? 1.0f : 0.0f;

        // conf term is needed everywhere (obj + noobj sums)
        const float bc = bcef(sigmoidf(base[4 * HW]), tconf[idx]);
        acc[4] += bc * m;
        acc[5] += bc * nm;
        acc[7] += m;
        acc[8] += nm;

        // x/y/w/h/cls terms are mask-gated: <=320 of 277k cells -> skip the
        // 80-channel logits and tcls reads entirely when mask==0.
        if (m != 0.0f) {
            const float s = bls[idx];
            acc[0] += bcef(sigmoidf(base[0]),      tx[idx]) * s;
            acc[1] += bcef(sigmoidf(base[HW]),     ty[idx]) * s;
            const float dw = base[2 * HW] - tw[idx];
            const float dh = base[3 * HW] - th[idx];
            acc[2] += dw * dw * s;
            acc[3] += dh * dh * s;
            const float* tc = tcls + (size_t)idx * NC;
            float sc = 0.0f;
#pragma unroll 4
            for (int c = 0; c < NC; ++c)
                sc += bcef(sigmoidf(base[(5 + c) * HW]), tc[c]);
            acc[6] += sc;
        }
    }

    // wave32 butterfly reduce
#pragma unroll
    for (int off = 16; off >= 1; off >>= 1)
#pragma unroll
        for (int j = 0; j < NACC; ++j)
            acc[j] += __shfl_xor(acc[j], off, 32);

    __shared__ float part[8][16];
    const int wave = threadIdx.x >> 5;
    const int lane = threadIdx.x & 31;
    if (lane == 0) {
#pragma unroll
        for (int j = 0; j < NACC; ++j) part[wave][j] = acc[j];
#pragma unroll
        for (int j = NACC; j < 16; ++j) part[wave][j] = 0.0f;
    }
    __syncthreads();

    if (threadIdx.x < 32) {                  // wave 0: EXEC all-ones within the wave
        v8f c = wmma_reduce8(part, lane);
        float* wsb = ws + (size_t)blockIdx.x * NACC;
        if (lane == 0) {                     // rows 0..7 (Sx..cnt)
#pragma unroll
            for (int j = 0; j < 8; ++j) wsb[j] = c[j];
        }
        if (lane == 16) {                    // row 8 (ncnt) = D[8][0]
            wsb[8] = c[0];
        }
    }
}

__global__ __launch_bounds__(128) void yolo_final_kernel(
    const float* __restrict__ ws, float* __restrict__ out)
{
    float acc[NACC];
#pragma unroll
    for (int j = 0; j < NACC; ++j) acc[j] = 0.0f;

    for (int e = threadIdx.x; e < NBLK; e += 128)
#pragma unroll
        for (int j = 0; j < NACC; ++j)
            acc[j] += ws[(size_t)e * NACC + j];

#pragma unroll
    for (int off = 16; off >= 1; off >>= 1)
#pragma unroll
        for (int j = 0; j < NACC; ++j)
            acc[j] += __shfl_xor(acc[j], off, 32);

    __shared__ float part[8][16];
    const int wave = threadIdx.x >> 5;
    const int lane = threadIdx.x & 31;
    if (lane == 0) {
#pragma unroll
        for (int j = 0; j < NACC; ++j) part[wave][j] = acc[j];
#pragma unroll
        for (int j = NACC; j < 16; ++j) part[wave][j] = 0.0f;
    }
    // only 4 waves here; zero the rest of the tile
    if (threadIdx.x < 32) {
#pragma unroll
        for (int w = 4; w < 8; ++w) part[w][lane & 15] = 0.0f;
    }
    __syncthreads();

    if (threadIdx.x < 32) {
        v8f c = wmma_reduce8(part, lane);
        // ncnt (row 8) lives in lane 16's c[0]; broadcast to the wave
        const float ncnt = __shfl(c[0], 16, 32);
        if (lane == 0) {
            const float Sx  = c[0], Sy = c[1], Sw = c[2], Sh = c[3];
            const float Scm = c[4], Scn = c[5], Scl = c[6], cnt = c[7];
            const float loss = 2.5f * (Sx + Sy + Sw + Sh) / cnt
                             + Scm / cnt + Scn / ncnt
                             + Scl / (cnt * (float)NC);
            out[0] = loss;
        }
    }
}

extern "C" void kernel_launch(void* const* d_in, const int* in_sizes, int n_in,
                              void* d_out, int out_size, void* d_ws, size_t ws_size,
                              hipStream_t stream) {
    const float*         in    = (const float*)d_in[0];
    const unsigned char* mask  = (const unsigned char*)d_in[1];   // jnp bool_
    const unsigned char* nmask = (const unsigned char*)d_in[2];   // jnp bool_
    const float*         tx    = (const float*)d_in[3];
    const float*         ty    = (const float*)d_in[4];
    const float*         tw    = (const float*)d_in[5];
    const float*         th    = (const float*)d_in[6];
    const float*         tconf = (const float*)d_in[7];
    const float*         tcls  = (const float*)d_in[8];
    const float*         bls   = (const float*)d_in[9];
    float* ws  = (float*)d_ws;      // NBLK * NACC floats (9 KB), rewritten each call
    float* out = (float*)d_out;

    yolo_partial_kernel<<<NBLK, NTHR, 0, stream>>>(in, mask, nmask, tx, ty, tw, th,
                                                   tconf, tcls, bls, ws);
    yolo_final_kernel<<<1, 128, 0, stream>>>(ws, out);
}